// MultiresEncoding_47785806135527
// MI455X (gfx1250) — compile-verified
//
#include <hip/hip_runtime.h>
#include <stdint.h>

// ---------------------------------------------------------------------------
// Multires hash-grid encoding (Instant-NGP style) for gfx1250 / MI455X.
//
// Roofline: output 256MB + coords 24MB streamed (NT hints, ~12us HBM floor at
// 23.3 TB/s); 30MB of hash tables are kept L2-resident (192MB L2) by marking
// all streaming traffic non-temporal so gathers never spill to HBM.
// Per-corner gather count halved by a feature-interleave prepass into d_ws.
// Dense level 0 (32KB) staged into LDS via gfx1250 async-to-LDS DMA.
// ---------------------------------------------------------------------------

typedef float v2f __attribute__((ext_vector_type(2)));
typedef float v4f __attribute__((ext_vector_type(4)));
typedef __attribute__((address_space(1))) int gint;  // global int
typedef __attribute__((address_space(3))) int lint;  // LDS int

namespace {
constexpr int      kLevels = 16;
constexpr int      kTable  = 1 << 18;           // 262144 entries per level
constexpr uint32_t kMask   = (uint32_t)(kTable - 1);
constexpr int      kR0     = 16;                // dense level resolution
constexpr int      kR0_3   = kR0 * kR0 * kR0;   // 4096
constexpr uint32_t kP1     = 2654435761u;
constexpr uint32_t kP2     = 805459861u;
// RES[l] = floor(16 * (512/16)^(l/15) + 1e-6)  (precomputed)
constexpr int kRes[kLevels] = {16, 20, 25, 32, 40, 50, 64, 80,
                               101, 128, 161, 203, 256, 322, 406, 512};
} // namespace

// Prepass: [15][2][T] -> [15][T][2] (v2f), NT reads (original layout is never
// touched again), regular stores so the interleaved tables become L2-resident.
__global__ __launch_bounds__(256) void interleave_tables_kernel(
    const float* __restrict__ t, v2f* __restrict__ o, int total) {
  const int i = blockIdx.x * 256 + threadIdx.x;
  if (i >= total) return;
  const int l = i >> 18;
  const int h = i & (int)kMask;
  const float* base = t + (size_t)l * 2 * kTable;
  v2f v;
  v.x = __builtin_nontemporal_load(base + h);
  v.y = __builtin_nontemporal_load(base + kTable + h);
  o[i] = v;
}

template <bool ILV>
__global__ __launch_bounds__(256) void encode_kernel(
    const float* __restrict__ coords,   // [N,3]
    const float* __restrict__ dense,    // [2,16,16,16]
    const float* __restrict__ tables,   // ILV: [15][T][2] ; else [15][2][T]
    float* __restrict__ out,            // [N,32]
    int N) {
  // Stage dense level-0 grid into LDS, feature-interleaved (32KB).
  __shared__ v2f sgrid[kR0_3];

#if __has_builtin(__builtin_amdgcn_global_load_async_to_lds_b32)
  for (int j = threadIdx.x; j < kR0_3; j += 256) {
    __builtin_amdgcn_global_load_async_to_lds_b32(
        (gint*)(dense + j), (lint*)(&sgrid[j]), 0, 0);
    __builtin_amdgcn_global_load_async_to_lds_b32(
        (gint*)(dense + kR0_3 + j), (lint*)((float*)&sgrid[j] + 1), 0, 0);
  }
#if __has_builtin(__builtin_amdgcn_s_wait_asynccnt)
  __builtin_amdgcn_s_wait_asynccnt(0);
#else
  asm volatile("s_wait_asynccnt 0x0" ::: "memory");
#endif
#else
  for (int j = threadIdx.x; j < kR0_3; j += 256) {
    v2f v;
    v.x = dense[j];
    v.y = dense[kR0_3 + j];
    sgrid[j] = v;
  }
#endif
  __syncthreads();

  const int i = blockIdx.x * 256 + threadIdx.x;
  if (i >= N) return;

  // Streamed once per thread -> non-temporal loads.
  const float cx = __builtin_nontemporal_load(coords + 3 * i + 0);
  const float cy = __builtin_nontemporal_load(coords + 3 * i + 1);
  const float cz = __builtin_nontemporal_load(coords + 3 * i + 2);
  // (c+1)*0.5 hoisted; per level *(res-1) matches reference expression tree.
  const float gx = (cx + 1.0f) * 0.5f;
  const float gy = (cy + 1.0f) * 0.5f;
  const float gz = (cz + 1.0f) * 0.5f;

  float* __restrict__ op = out + (size_t)i * 32;
  float p0 = 0.0f, p1 = 0.0f;  // previous level's features (for b128 packing)

#pragma unroll
  for (int l = 0; l < kLevels; ++l) {
    const int   res = kRes[l];
    const float rm1 = (float)(res - 1);

    const float px = fminf(fmaxf(gx * rm1, 0.0f), rm1);
    const float py = fminf(fmaxf(gy * rm1, 0.0f), rm1);
    const float pz = fminf(fmaxf(gz * rm1, 0.0f), rm1);

    const float fx = floorf(px), fy = floorf(py), fz = floorf(pz);
    const float wx = px - fx, wy = py - fy, wz = pz - fz;
    const float wx0 = 1.0f - wx, wy0 = 1.0f - wy, wz0 = 1.0f - wz;

    const int x0 = (int)fx, y0 = (int)fy, z0 = (int)fz;
    const int x1 = min(x0 + 1, res - 1);
    const int y1 = min(y0 + 1, res - 1);
    const int z1 = min(z0 + 1, res - 1);

    // Factored corner weights: 4 + 8 mults instead of 16.
    // Corner bit layout (matches reference): bit2 = x, bit1 = y, bit0 = z.
    // Product order matches reference reduction: (sx*sy)*sz.
    const float wzc[2] = {wz0, wz};
    const float wxy[4] = {wx0 * wy0, wx0 * wy, wx * wy0, wx * wy};

    float a0 = 0.0f, a1 = 0.0f;

    if (l == 0) {
      // Dense level: v2f gathers from LDS (ds_load_b64).
      const int bxy[4] = {(x0 * kR0 + y0) * kR0, (x0 * kR0 + y1) * kR0,
                          (x1 * kR0 + y0) * kR0, (x1 * kR0 + y1) * kR0};
      const int zz[2] = {z0, z1};
#pragma unroll
      for (int c = 0; c < 8; ++c) {
        const float w = wxy[c >> 1] * wzc[c & 1];
        const v2f g = sgrid[bxy[c >> 1] + zz[c & 1]];
        a0 += g.x * w;
        a1 += g.y * w;
      }
    } else {
      // Hash levels: prime 0 is 1 -> 4 multiplies, then factored XORs.
      const uint32_t hy0 = (uint32_t)y0 * kP1, hy1 = (uint32_t)y1 * kP1;
      const uint32_t hz[2]  = {(uint32_t)z0 * kP2, (uint32_t)z1 * kP2};
      const uint32_t hxy[4] = {(uint32_t)x0 ^ hy0, (uint32_t)x0 ^ hy1,
                               (uint32_t)x1 ^ hy0, (uint32_t)x1 ^ hy1};
#pragma unroll
      for (int c = 0; c < 8; ++c) {
        const uint32_t h = (hxy[c >> 1] ^ hz[c & 1]) & kMask;
        const float w = wxy[c >> 1] * wzc[c & 1];
        float g0, g1;
        if (ILV) {
          const v2f g = ((const v2f*)tables)[(size_t)(l - 1) * kTable + h];
          g0 = g.x;
          g1 = g.y;
        } else {
          const float* base = tables + (size_t)(l - 1) * 2 * kTable;
          g0 = base[h];
          g1 = base[kTable + h];
        }
        a0 += g0 * w;
        a1 += g1 * w;
      }
    }

    if (l & 1) {
      // Pack two levels -> one NT b128 store (don't evict tables from L2).
      v4f v = {p0, p1, a0, a1};
      __builtin_nontemporal_store(v, (v4f*)op + (l >> 1));
    } else {
      p0 = a0;
      p1 = a1;
    }
  }
}

extern "C" void kernel_launch(void* const* d_in, const int* in_sizes, int n_in,
                              void* d_out, int out_size, void* d_ws,
                              size_t ws_size, hipStream_t stream) {
  const float* coords = (const float*)d_in[0];  // [N,3]
  const float* dense  = (const float*)d_in[1];  // [2,16,16,16]
  const float* tables = (const float*)d_in[2];  // [15,2,T]
  float* out = (float*)d_out;                   // [N,32]

  const int N = in_sizes[0] / 3;
  const int blocks = (N + 255) / 256;

  const size_t need = (size_t)(kLevels - 1) * kTable * 2 * sizeof(float);
  if (d_ws != nullptr && ws_size >= need) {
    const int total = (kLevels - 1) * kTable;
    interleave_tables_kernel<<<(total + 255) / 256, 256, 0, stream>>>(
        tables, (v2f*)d_ws, total);
    encode_kernel<true><<<blocks, 256, 0, stream>>>(
        coords, dense, (const float*)d_ws, out, N);
  } else {
    encode_kernel<false><<<blocks, 256, 0, stream>>>(coords, dense, tables,
                                                     out, N);
  }
}